// ProtoPNet_29549374996490
// MI455X (gfx1250) — compile-verified
//
#include <hip/hip_runtime.h>
#include <hip/hip_bf16.h>
#include <math.h>

// Problem constants
#define BATCH    64
#define CIN      512
#define HW       196          // 14*14
#define NPROTO   2000
#define NCLS     200
#define EPSV     0.0001f

// Output slice offsets (floats), concatenated in reference return order:
// (x, min_dists, dists, activations, logits)
#define OFF_X       0
#define N_X         (BATCH * CIN * HW)               // 6,422,528
#define OFF_MIN     (OFF_X + N_X)                    // 6,422,528
#define N_MIN       (BATCH * NPROTO)                 // 128,000
#define OFF_DISTS   (OFF_MIN + N_MIN)                // 6,550,528
#define N_DISTS     (BATCH * NPROTO * HW)            // 25,088,000
#define OFF_ACT     (OFF_DISTS + N_DISTS)            // 31,638,528
#define N_ACT       (BATCH * NPROTO)                 // 128,000
#define OFF_LOGITS  (OFF_ACT + N_ACT)                // 31,766,528

// Tiling: 16(M=proto) x 32(N=hw) per wave, two v8f accumulators.
#define PT_TILES  (NPROTO / 16)                      // 125
#define HW_TILES2 ((HW + 31) / 32)                   // 7 (covers 224, clamped)
#define TOT_TILES (BATCH * PT_TILES * HW_TILES2)     // 56,000
#define WAVES_PER_BLOCK 8
#define DISTS_GRID (TOT_TILES / WAVES_PER_BLOCK)     // 7,000 exactly

typedef __attribute__((ext_vector_type(2))) float v2f;
typedef __attribute__((ext_vector_type(8))) float v8f;

__device__ __forceinline__ int imin(int a, int b) { return a < b ? a : b; }

// ds_swizzle group-of-32 XOR swap: imm = (xor_mask<<10) | (or=0)<<5 | and_mask 0x1f
#define SWZ_MIN(v, imm) \
    fminf((v), __int_as_float(__builtin_amdgcn_ds_swizzle(__float_as_int(v), (imm))))

// ---------------------------------------------------------------------------
// Prep: p2 sums, x2 patch sums, +INF init of min_dists (as raw uint bits).
// grid 500 x 256 == 128,000 threads (covers the largest range).
// ---------------------------------------------------------------------------
__global__ void k_prep(const float* __restrict__ x,
                       const float* __restrict__ proto,
                       float* __restrict__ p2,
                       float* __restrict__ x2,
                       unsigned* __restrict__ min_bits) {
    const int i = blockIdx.x * blockDim.x + threadIdx.x;

    if (i < NPROTO) {
        const float4* row = (const float4*)(proto + (size_t)i * CIN);
        float s = 0.f;
        #pragma unroll 4
        for (int c = 0; c < CIN / 4; ++c) {
            float4 v = row[c];
            s += v.x * v.x + v.y * v.y + v.z * v.z + v.w * v.w;
        }
        p2[i] = s;
    }

    if (i < BATCH * HW) {
        const int b = i / HW, hw = i % HW;
        const float* xb = x + (size_t)b * CIN * HW + hw;
        float s = 0.f;
        #pragma unroll 8
        for (int c = 0; c < CIN; ++c) {
            float v = xb[(size_t)c * HW];
            s += v * v;
        }
        x2[i] = s;
    }

    if (i < BATCH * NPROTO) {
        min_bits[i] = 0x7F800000u;   // +INF bit pattern
    }
}

// ---------------------------------------------------------------------------
// Passthrough copy of x into output slot 0 (float4 granularity).
// ---------------------------------------------------------------------------
__global__ void k_copy_x(const float4* __restrict__ src, float4* __restrict__ dst) {
    const int i = blockIdx.x * blockDim.x + threadIdx.x;
    if (i < N_X / 4) dst[i] = src[i];
}

// ---------------------------------------------------------------------------
// Main fused kernel: xp GEMM via V_WMMA_F32_16X16X4_F32 with 16x32 register
// blocking (A float2 shared across both N-half WMMAs). dists = relu(...),
// store both tiles, ds_swizzle XOR-min within 16-lane groups, one atomicMin
// per C/D row-half. One wave per (b, p-tile, hw-pair-tile).
// ---------------------------------------------------------------------------
__global__ void k_dists(const float* __restrict__ x,
                        const float* __restrict__ proto,
                        const float* __restrict__ p2,
                        const float* __restrict__ x2,
                        float* __restrict__ out) {
    const int lane  = threadIdx.x & 31;
    const int wid   = threadIdx.x >> 5;
    const int tile  = blockIdx.x * WAVES_PER_BLOCK + wid;

    const int b   = tile / (PT_TILES * HW_TILES2);
    const int rem = tile % (PT_TILES * HW_TILES2);
    const int p0  = (rem / HW_TILES2) * 16;
    const int hw0 = (rem % HW_TILES2) * 32;

    const int mrow  = lane & 15;      // A row
    const int khalf = lane >> 4;      // 0: K0/K1 half, 1: K2/K3 half
    const int ncol  = lane & 15;      // B/C column within a 16-wide tile
    const int hwc0  = imin(hw0 + ncol,      HW - 1);   // clamped (EXEC stays full)
    const int hwc1  = imin(hw0 + 16 + ncol, HW - 1);
    const bool valid0 = (hw0 + ncol)      < HW;
    const bool valid1 = (hw0 + 16 + ncol) < HW;

    const float* __restrict__ aptr  = proto + (size_t)(p0 + mrow) * CIN + khalf * 2;
    const float* __restrict__ xb    = x + (size_t)b * CIN * HW + (size_t)(khalf * 2) * HW;
    const float* __restrict__ bptr0 = xb + hwc0;
    const float* __restrict__ bptr1 = xb + hwc1;

    __builtin_prefetch(aptr, 0, 0);
    __builtin_prefetch(bptr0, 0, 0);

    v8f acc0 = {};
    v8f acc1 = {};
    #pragma unroll 4
    for (int k = 0; k < CIN; k += 4) {
        // A (16x4 fp32): lane<16 -> {K=k, K=k+1}; lane>=16 -> {K=k+2, K=k+3}
        v2f a = *(const v2f*)(aptr + k);
        // B (4x16 fp32), two N-halves sharing A
        v2f b0, b1;
        b0.x = bptr0[(size_t)k * HW];
        b0.y = bptr0[(size_t)k * HW + HW];
        b1.x = bptr1[(size_t)k * HW];
        b1.y = bptr1[(size_t)k * HW + HW];
        acc0 = __builtin_amdgcn_wmma_f32_16x16x4_f32(
                   false, a, false, b0, (short)0, acc0, false, false);
        acc1 = __builtin_amdgcn_wmma_f32_16x16x4_f32(
                   false, a, false, b1, (short)0, acc1, false, false);
    }

    const float x2v0 = x2[b * HW + hwc0];
    const float x2v1 = x2[b * HW + hwc1];
    float* __restrict__ dists = out + OFF_DISTS + (size_t)b * NPROTO * HW;
    unsigned* __restrict__ minb = (unsigned*)(out + OFF_MIN) + b * NPROTO;

    #pragma unroll
    for (int r = 0; r < 8; ++r) {
        const int p = p0 + r + khalf * 8;      // C/D layout: VGPR r -> M=r (+8 hi half)
        float dv0 = x2v0 - 2.0f * acc0[r] + p2[p];
        float dv1 = x2v1 - 2.0f * acc1[r] + p2[p];
        dv0 = dv0 > 0.0f ? dv0 : 0.0f;
        dv1 = dv1 > 0.0f ? dv1 : 0.0f;
        if (valid0) dists[(size_t)p * HW + hw0 + ncol]      = dv0;
        if (valid1) dists[(size_t)p * HW + hw0 + 16 + ncol] = dv1;

        // fold both N-halves, then XOR-min across the 16-lane group (ds_swizzle)
        float mv = fminf(valid0 ? dv0 : __builtin_inff(),
                         valid1 ? dv1 : __builtin_inff());
        mv = SWZ_MIN(mv, 0x041F);   // xor 1
        mv = SWZ_MIN(mv, 0x081F);   // xor 2
        mv = SWZ_MIN(mv, 0x101F);   // xor 4
        mv = SWZ_MIN(mv, 0x201F);   // xor 8
        if (ncol == 0) atomicMin(minb + p, __float_as_uint(mv));  // non-neg: uint order == float order
    }
}

// ---------------------------------------------------------------------------
// Final: activations = log((m+1)/(m+eps)); logits = act @ fc_weight^T.
// One block per batch element.
// ---------------------------------------------------------------------------
__global__ void k_logits(const float* __restrict__ fcw, float* __restrict__ out) {
    __shared__ float sact[NPROTO];
    const int b = blockIdx.x;
    const int tid = threadIdx.x;

    const float* __restrict__ md = out + OFF_MIN + (size_t)b * NPROTO;
    float* __restrict__ act = out + OFF_ACT + (size_t)b * NPROTO;

    for (int p = tid; p < NPROTO; p += blockDim.x) {
        const float m = md[p];
        const float a = logf((m + 1.0f) / (m + EPSV));
        sact[p] = a;
        act[p] = a;
    }
    __syncthreads();

    float* __restrict__ lg = out + OFF_LOGITS + (size_t)b * NCLS;
    for (int c = tid; c < NCLS; c += blockDim.x) {
        const float* __restrict__ w = fcw + (size_t)c * NPROTO;
        float s = 0.f;
        #pragma unroll 4
        for (int p = 0; p < NPROTO; ++p) s += sact[p] * w[p];
        lg[c] = s;
    }
}

// ---------------------------------------------------------------------------
extern "C" void kernel_launch(void* const* d_in, const int* in_sizes, int n_in,
                              void* d_out, int out_size, void* d_ws, size_t ws_size,
                              hipStream_t stream) {
    const float* x     = (const float*)d_in[0];   // (64, 512, 14, 14)
    const float* proto = (const float*)d_in[1];   // (2000, 512, 1, 1)
    const float* fcw   = (const float*)d_in[2];   // (200, 2000)
    float* out = (float*)d_out;

    float* p2 = (float*)d_ws;                     // 2000 floats
    float* x2 = (float*)d_ws + 4096;              // 64*196 = 12544 floats

    k_prep<<<500, 256, 0, stream>>>(x, proto, p2, x2,
                                    (unsigned*)(out + OFF_MIN));
    k_copy_x<<<(N_X / 4 + 255) / 256, 256, 0, stream>>>((const float4*)x,
                                                        (float4*)(out + OFF_X));
    k_dists<<<DISTS_GRID, 256, 0, stream>>>(x, proto, p2, x2, out);
    k_logits<<<BATCH, 256, 0, stream>>>(fcw, out);
}